// SelectiveDense_39092792328192
// MI455X (gfx1250) — compile-verified
//
#include <hip/hip_runtime.h>

typedef __attribute__((ext_vector_type(2))) float v2f;
typedef __attribute__((ext_vector_type(8))) float v8f;

#define B_DIM 8192
#define D_DIM 4096
#define G_DIM 256
#define K_DIM 32
#define U_DIM 64

// LDS strides (in floats) chosen for conflict-free WMMA fragment reads.
#define SEL_STRIDE 36   // 36*4B row stride; 16B-aligned float4 stores; banks disjoint per half-wave
#define KER_STRIDE 80   // +2 rows => +32 banks, so half-waves never collide

__global__ __launch_bounds__(256) void selective_dense_wmma(
    const float* __restrict__ x,        // [B, D]
    const float* __restrict__ kernels,  // [G, K, U]
    const float* __restrict__ bias,     // [U]
    const int*   __restrict__ indices,  // [G, K]
    float*       __restrict__ out)      // [B, G, U]
{
    __shared__ float selLDS[32 * SEL_STRIDE];   // 32 b-rows x 32 k
    __shared__ float kerLDS[32 * KER_STRIDE];   // 32 k x 64 u

    const int g    = blockIdx.y;
    const int bblk = blockIdx.x;                // 32 rows of B per block
    const int t    = threadIdx.x;
    const int lane = t & 31;
    const int wave = t >> 5;

    // ---- stage kernels[g] (32x64 f32 = 8KB) into LDS: 256 thr x 8 floats ----
    {
        const int r  = t >> 3;                  // k-row 0..31
        const int c0 = (t & 7) * 8;             // u-col 0,8,...,56
        const float4* src =
            (const float4*)(kernels + (size_t)g * (K_DIM * U_DIM) + r * U_DIM + c0);
        float4 v0 = src[0];
        float4 v1 = src[1];
        float* dst = &kerLDS[r * KER_STRIDE + c0];
        *(float4*)(dst)     = v0;
        *(float4*)(dst + 4) = v1;
    }

    // ---- gather sel tile (32 b-rows x 32 k) into LDS: 256 thr x 4 gathers ----
    {
        const int bl = t >> 3;                  // local b-row 0..31
        const int k0 = (t & 7) * 4;             // k 0,4,...,28
        const int4 idx4 = *(const int4*)(indices + g * K_DIM + k0);
        const float* xrow = x + (size_t)(bblk * 32 + bl) * D_DIM;
        float4 s;
        s.x = xrow[idx4.x];
        s.y = xrow[idx4.y];
        s.z = xrow[idx4.z];
        s.w = xrow[idx4.w];
        *(float4*)&selLDS[bl * SEL_STRIDE + k0] = s;
    }

    __syncthreads();

    // 8 waves: 2 b-tiles x 4 n-tiles of 16x16 output
    const int bt    = wave >> 2;                // 0..1
    const int ntile = wave & 3;                 // 0..3
    const int half  = lane >> 4;                // 0 = lanes 0-15, 1 = lanes 16-31
    const int l15   = lane & 15;

    v8f acc = {};
    const float* aRow = &selLDS[(bt * 16 + l15) * SEL_STRIDE];   // A: row m = l15
    const float* bCol = &kerLDS[ntile * 16 + l15];               // B: col n = l15

    // K=32 as 8 chained V_WMMA_F32_16X16X4_F32
    #pragma unroll
    for (int kk = 0; kk < 8; ++kk) {
        const int ka = kk * 4 + half * 2;       // K index for VGPR0; VGPR1 = ka+1
        v2f a, b;
        a.x = aRow[ka];
        a.y = aRow[ka + 1];
        b.x = bCol[(size_t)ka * KER_STRIDE];
        b.y = bCol[(size_t)(ka + 1) * KER_STRIDE];
        acc = __builtin_amdgcn_wmma_f32_16x16x4_f32(
            /*neg_a=*/false, a, /*neg_b=*/false, b,
            /*c_mod=*/(short)0, acc, /*reuse_a=*/false, /*reuse_b=*/false);
    }

    // Epilogue: + bias, ReLU, store. C/D layout: VGPR r -> m = r + 8*half, n = l15.
    const float bn = bias[ntile * 16 + l15];
    const size_t rowStride = (size_t)G_DIM * U_DIM;
    const size_t outBase = (size_t)(bblk * 32 + bt * 16 + half * 8) * rowStride
                         + (size_t)g * U_DIM + ntile * 16 + l15;
    #pragma unroll
    for (int r = 0; r < 8; ++r) {
        float v = acc[r] + bn;
        v = v > 0.0f ? v : 0.0f;
        out[outBase + (size_t)r * rowStride] = v;
    }
}

extern "C" void kernel_launch(void* const* d_in, const int* in_sizes, int n_in,
                              void* d_out, int out_size, void* d_ws, size_t ws_size,
                              hipStream_t stream) {
    const float* x       = (const float*)d_in[0];
    const float* kernels = (const float*)d_in[1];
    const float* bias    = (const float*)d_in[2];
    const int*   indices = (const int*)d_in[3];
    float*       out     = (float*)d_out;

    dim3 grid(B_DIM / 32, G_DIM);   // 256 x 256 workgroups
    dim3 block(256);                // 8 wave32
    selective_dense_wmma<<<grid, block, 0, stream>>>(x, kernels, bias, indices, out);
}